// SegmentCausalCrossAttention_86620900426242
// MI455X (gfx1250) — compile-verified
//
#include <hip/hip_runtime.h>
#include <cfloat>

// ---------------------------------------------------------------------------
// Problem constants (from reference)
// ---------------------------------------------------------------------------
#define B_    4
#define LQ_   4096
#define LKV_  512
#define DATTN 1024
#define NH    16
#define HD    64
#define KW_   4          // LOOKBACK + 1

typedef __attribute__((ext_vector_type(2))) float v2f;
typedef __attribute__((ext_vector_type(8))) float v8f;
typedef __attribute__((ext_vector_type(4))) int   v4i;

typedef __attribute__((address_space(1))) v4i* gv4i_p;   // global
typedef __attribute__((address_space(3))) v4i* lv4i_p;   // LDS

// ---------------------------------------------------------------------------
// Async global->LDS staging (gfx1250). Falls back to a synchronous copy if
// the builtins are not declared by this toolchain.
// ---------------------------------------------------------------------------
#if __has_builtin(__builtin_amdgcn_global_load_async_to_lds_b128)
#define HAS_ASYNC_LDS 1
#endif

__device__ __forceinline__ void stage16(const float* g, float* l) {
#ifdef HAS_ASYNC_LDS
  __builtin_amdgcn_global_load_async_to_lds_b128((gv4i_p)g, (lv4i_p)l, 0, 0);
#else
  *(float4*)l = *(const float4*)g;
#endif
}

__device__ __forceinline__ void async_wait_all() {
#ifdef HAS_ASYNC_LDS
#if __has_builtin(__builtin_amdgcn_s_wait_asynccnt)
  __builtin_amdgcn_s_wait_asynccnt(0);
#else
  asm volatile("s_wait_asynccnt 0x0" ::: "memory");
#endif
#endif
}

// ---------------------------------------------------------------------------
// Y[m][n] = sum_k X[m][k] * W[n][k]   (torch-style y = x @ W.T)
// X: M x K row-major, W: N x K row-major, Y: M x N row-major.
// M, N multiples of 128; K multiple of 16.
//
// Workgroup (256 thr = 8 waves) computes a 128x128 output block.
// Wave w: M-strip (w>>1)*32 (two 16-row tiles), N-strip (w&1)*64 (four tiles).
// K is consumed in chunks of 16, double-buffered in LDS via async loads.
//
// LDS row stride = 20 floats (16 data + 4 pad): 20*r mod 64 distinct and
// multiple of 4 for r=0..15 -> conflict-free ds_load_b64 fragment reads,
// and 16B-aligned rows for the async b128 writes.
//
// V_WMMA_F32_16X16X4_F32 operand layout (ISA 7.12.2):
//   A 16x4 : lane = M + 16*Khalf, holds K = 2*Khalf + {0,1}  -> float2/lane
//   B 4x16 : lane = N + 16*Khalf, holds K = 2*Khalf + {0,1}  -> float2/lane
//   C/D    : VGPR v, lanes 0-15 -> row v, lanes 16-31 -> row v+8, col=lane&15
// ---------------------------------------------------------------------------
#define KC   16
#define LDP  20          // padded LDS row stride (floats)

__global__ __launch_bounds__(256) void gemm_xwT_wmma(
    const float* __restrict__ X, const float* __restrict__ W,
    float* __restrict__ Y, int M, int N, int K)
{
  __shared__ float Xs[2][128 * LDP];
  __shared__ float Ws[2][128 * LDP];

  const int t     = threadIdx.x;
  const int lane  = t & 31;
  const int wave  = t >> 5;
  const int mBlk  = blockIdx.y * 128;
  const int nBlk  = blockIdx.x * 128;
  const int mw    = (wave >> 1) * 32;   // wave M offset within block
  const int nw    = (wave & 1) * 64;    // wave N offset within block
  const int idx16 = lane & 15;
  const int kh    = lane >> 4;          // K-half: 0 -> K{0,1}, 1 -> K{2,3}

  // Staging map: per chunk, each matrix moves 128 rows x 16 floats
  // = 512 x b128 transfers; 256 threads issue 2 each per matrix.
  const int sRow0 = t >> 2;             // lin = 0*256 + t
  const int sCg0  = (t & 3) * 4;
  const int sRow1 = (256 + t) >> 2;     // lin = 1*256 + t
  const int sCg1  = ((256 + t) & 3) * 4;

  v8f acc[8] = {};                      // [mt*4 + nt]

  // ---- prologue: stage chunk 0 into buffer 0 --------------------------------
  {
    stage16(X + (size_t)(mBlk + sRow0) * K + sCg0, &Xs[0][sRow0 * LDP + sCg0]);
    stage16(X + (size_t)(mBlk + sRow1) * K + sCg1, &Xs[0][sRow1 * LDP + sCg1]);
    stage16(W + (size_t)(nBlk + sRow0) * K + sCg0, &Ws[0][sRow0 * LDP + sCg0]);
    stage16(W + (size_t)(nBlk + sRow1) * K + sCg1, &Ws[0][sRow1 * LDP + sCg1]);
  }
  async_wait_all();
  __syncthreads();

  for (int kc = 0; kc < K; kc += KC) {
    const int buf = (kc >> 4) & 1;

    // ---- stage next chunk into the other buffer (overlaps with compute) ----
    if (kc + KC < K) {
      const int kn = kc + KC;
      stage16(X + (size_t)(mBlk + sRow0) * K + kn + sCg0, &Xs[buf ^ 1][sRow0 * LDP + sCg0]);
      stage16(X + (size_t)(mBlk + sRow1) * K + kn + sCg1, &Xs[buf ^ 1][sRow1 * LDP + sCg1]);
      stage16(W + (size_t)(nBlk + sRow0) * K + kn + sCg0, &Ws[buf ^ 1][sRow0 * LDP + sCg0]);
      stage16(W + (size_t)(nBlk + sRow1) * K + kn + sCg1, &Ws[buf ^ 1][sRow1 * LDP + sCg1]);
    }

    // ---- compute current chunk from LDS ------------------------------------
#pragma unroll
    for (int ks = 0; ks < KC; ks += 4) {
      const int kb = ks + 2 * kh;
      const v2f a0 = *(const v2f*)&Xs[buf][(mw + idx16)      * LDP + kb];
      const v2f a1 = *(const v2f*)&Xs[buf][(mw + 16 + idx16) * LDP + kb];
#pragma unroll
      for (int nt = 0; nt < 4; ++nt) {
        const v2f b = *(const v2f*)&Ws[buf][(nw + nt * 16 + idx16) * LDP + kb];
        acc[nt]     = __builtin_amdgcn_wmma_f32_16x16x4_f32(false, a0, false, b, (short)0, acc[nt],     false, false);
        acc[4 + nt] = __builtin_amdgcn_wmma_f32_16x16x4_f32(false, a1, false, b, (short)0, acc[4 + nt], false, false);
      }
    }

    async_wait_all();     // next chunk resident
    __syncthreads();      // all waves done reading buf / writing buf^1
  }

  // ---- write back -----------------------------------------------------------
#pragma unroll
  for (int mt = 0; mt < 2; ++mt) {
#pragma unroll
    for (int nt = 0; nt < 4; ++nt) {
      float* yb = Y + (size_t)(mBlk + mw + mt * 16 + kh * 8) * N
                    + nBlk + nw + nt * 16 + idx16;
      const v8f a = acc[mt * 4 + nt];
#pragma unroll
      for (int v = 0; v < 8; ++v) yb[(size_t)v * N] = a[v];
    }
  }
}

// ---------------------------------------------------------------------------
// Windowed segment attention. One thread = one (b, q, h).
//  qh : (B, LQ, DATTN)      kv : (B, LKV, 2*DATTN)  [k | v]
//  For w in 0..3: idx = seg_id[b,q] - w; idx < 0 is masked (-inf score).
// ---------------------------------------------------------------------------
__global__ __launch_bounds__(256) void window_attn(
    const float* __restrict__ qh, const float* __restrict__ kv,
    const int* __restrict__ seg, float* __restrict__ out)
{
  const int t = blockIdx.x * blockDim.x + threadIdx.x;
  const int h = t & (NH - 1);
  const int q = (t >> 4) & (LQ_ - 1);
  const int b = t >> 16;                      // 16 = log2(NH * LQ_)
  if (b >= B_) return;

  const int s0 = seg[b * LQ_ + q];
  const float scale = 0.125f;                 // 64^-0.5

  const float* qv = qh + ((size_t)b * LQ_ + q) * DATTN + h * HD;
  float4 qr[HD / 4];
#pragma unroll
  for (int c = 0; c < HD / 4; ++c) qr[c] = ((const float4*)qv)[c];

  float sc[KW_];
  int   valid[KW_];
#pragma unroll
  for (int w = 0; w < KW_; ++w) {
    const int idx = s0 - w;
    valid[w] = (idx >= 0);
    float s = 0.f;
    if (valid[w]) {
      const float* kr = kv + ((size_t)b * LKV_ + idx) * (2 * DATTN) + h * HD;
#pragma unroll
      for (int c = 0; c < HD / 4; ++c) {
        const float4 kk = ((const float4*)kr)[c];
        s += qr[c].x * kk.x + qr[c].y * kk.y + qr[c].z * kk.z + qr[c].w * kk.w;
      }
    }
    sc[w] = s * scale;
  }

  float mx = -FLT_MAX;
#pragma unroll
  for (int w = 0; w < KW_; ++w) if (valid[w] && sc[w] > mx) mx = sc[w];
  float p[KW_];
  float sum = 0.f;
#pragma unroll
  for (int w = 0; w < KW_; ++w) {
    p[w] = valid[w] ? __expf(sc[w] - mx) : 0.f;
    sum += p[w];
  }
  const float inv = 1.f / sum;
#pragma unroll
  for (int w = 0; w < KW_; ++w) p[w] *= inv;

  float* ov = out + ((size_t)b * LQ_ + q) * DATTN + h * HD;
#pragma unroll
  for (int c = 0; c < HD / 4; ++c) {
    float4 a = make_float4(0.f, 0.f, 0.f, 0.f);
#pragma unroll
    for (int w = 0; w < KW_; ++w) {
      if (valid[w]) {
        const float* vr = kv + ((size_t)b * LKV_ + (s0 - w)) * (2 * DATTN) + DATTN + h * HD;
        const float4 vv = ((const float4*)vr)[c];
        a.x += p[w] * vv.x; a.y += p[w] * vv.y;
        a.z += p[w] * vv.z; a.w += p[w] * vv.w;
      }
    }
    ((float4*)ov)[c] = a;
  }
}

// ---------------------------------------------------------------------------
// Launcher. Inputs (setup_inputs order): q, kv_src, seg_id, Wq, Wkv, Wo.
// Workspace layout: qh (64MB) | kv (16MB) | attn (64MB)  = 144MB.
// ---------------------------------------------------------------------------
extern "C" void kernel_launch(void* const* d_in, const int* in_sizes, int n_in,
                              void* d_out, int out_size, void* d_ws, size_t ws_size,
                              hipStream_t stream) {
  const float* q      = (const float*)d_in[0];
  const float* kv_src = (const float*)d_in[1];
  const int*   seg    = (const int*)d_in[2];
  const float* Wq     = (const float*)d_in[3];
  const float* Wkv    = (const float*)d_in[4];
  const float* Wo     = (const float*)d_in[5];
  float* out = (float*)d_out;

  float* qh   = (float*)d_ws;                               // B*LQ*DATTN
  float* kv   = qh + (size_t)B_ * LQ_ * DATTN;              // B*LKV*2*DATTN
  float* attn = kv + (size_t)B_ * LKV_ * 2 * DATTN;         // B*LQ*DATTN

  const dim3 blk(256);

  // qh = q @ Wq.T                    M=16384 N=1024 K=1024
  gemm_xwT_wmma<<<dim3(1024 / 128, (B_ * LQ_) / 128), blk, 0, stream>>>(
      q, Wq, qh, B_ * LQ_, DATTN, 1024);

  // kv = kv_src @ Wkv.T              M=2048  N=2048 K=1024
  gemm_xwT_wmma<<<dim3((2 * DATTN) / 128, (B_ * LKV_) / 128), blk, 0, stream>>>(
      kv_src, Wkv, kv, B_ * LKV_, 2 * DATTN, 1024);

  // windowed attention               (B*LQ*NH) threads
  window_attn<<<dim3((B_ * LQ_ * NH) / 256), dim3(256), 0, stream>>>(
      qh, kv, seg, attn);

  // out = attn @ Wo.T                M=16384 N=1024 K=1024
  gemm_xwT_wmma<<<dim3(1024 / 128, (B_ * LQ_) / 128), blk, 0, stream>>>(
      attn, Wo, out, B_ * LQ_, DATTN, 1024);
}